// Net_25598005085127
// MI455X (gfx1250) — compile-verified
//
#include <hip/hip_runtime.h>

typedef float v2f __attribute__((ext_vector_type(2)));
typedef float v8f __attribute__((ext_vector_type(8)));

// Problem shape (from the reference): out[b] = U[b]^T @ x[b]
constexpr int Bn = 128;   // graphs
constexpr int Nn = 1024;  // nodes per graph (contraction dim)
constexpr int Mn = 512;   // pooled nodes per graph
constexpr int Fn = 32;    // features

constexpr int KC = 128;   // K rows of x staged in LDS per chunk
// LDS layout: element (n, f) lives at dword (n>>1)*RP + 2*f + (n&1).
// RP = 96 (64 data dwords + 32 pad) => the two half-wave groups of a b64 read
// (row-pairs kk/2 and kk/2+1) differ by 96 ≡ 32 (mod 64) banks: conflict-free.
constexpr int RP = 96;

#if defined(__has_builtin)
#if __has_builtin(__builtin_amdgcn_global_load_async_to_lds_b32)
#define HAVE_ASYNC_LDS 1
#endif
#endif

#define GAS __attribute__((address_space(1)))
#define LAS __attribute__((address_space(3)))

__global__ __launch_bounds__(256) void haarpool_wmma_f32(
    const float* __restrict__ x,   // [B*N, F]
    const float* __restrict__ U,   // [B, N, M]
    float* __restrict__ out)       // [B*M, F]
{
  __shared__ float xs[(KC / 2) * RP];  // 24 KB, row-pair interleaved

  const int b      = blockIdx.x >> 2;   // graph index
  const int mchunk = blockIdx.x & 3;    // which 128-row chunk of M
  const int tid    = threadIdx.x;
  const int wave   = tid >> 5;          // 8 waves per WG
  const int lane   = tid & 31;
  const int lo     = lane & 15;
  const int hi     = lane >> 4;

  const float* __restrict__ Ub = U + (size_t)b * Nn * Mn;
  const float* __restrict__ xb = x + (size_t)b * Nn * Fn;
  const int m0 = mchunk * 128 + wave * 16;

  // staging role: lane owns feature column f and row-pair family rp0
  const int f   = tid & 31;
  const int rp0 = tid >> 5;  // 0..7

  v8f acc0 = {};  // f = 0..15
  v8f acc1 = {};  // f = 16..31

  for (int k0 = 0; k0 < Nn; k0 += KC) {
    __syncthreads();  // previous chunk's readers done before overwrite

    // ---- stage x[b][k0:k0+KC][0:32] into LDS (row-pair interleaved) ----
#if HAVE_ASYNC_LDS
    {
      const float* gb = xb + (size_t)(k0 + 2 * rp0) * Fn + f;
      float*       lb = &xs[rp0 * RP + 2 * f];
#pragma unroll
      for (int i = 0; i < 8; ++i) {
        // (n = k0 + 2*(rp0+8i), f)  -> lds dword +0
        __builtin_amdgcn_global_load_async_to_lds_b32(
            (GAS int*)(gb + (size_t)i * 16 * Fn),
            (LAS int*)(lb + i * 8 * RP), 0, 0);
        // (n+1, f)                  -> lds dword +1
        __builtin_amdgcn_global_load_async_to_lds_b32(
            (GAS int*)(gb + (size_t)i * 16 * Fn + Fn),
            (LAS int*)(lb + i * 8 * RP + 1), 0, 0);
      }
#if __has_builtin(__builtin_amdgcn_s_wait_asynccnt)
      __builtin_amdgcn_s_wait_asynccnt(0);
#else
      asm volatile("s_wait_asynccnt 0x0" ::: "memory");
#endif
    }
#else
    {
#pragma unroll
      for (int i = 0; i < 8; ++i) {
        const int rp = rp0 + 8 * i;
        const int n  = k0 + 2 * rp;
        v2f v;
        v.x = xb[(size_t)n * Fn + f];
        v.y = xb[(size_t)(n + 1) * Fn + f];
        *(v2f*)&xs[rp * RP + 2 * f] = v;  // b64 store, conflict-free
      }
    }
#endif
    __syncthreads();

    // ---- compute: A fragment rows K = kk + 2*hi (+1), col m0+lo ----
    const float* __restrict__ Ua = Ub + (size_t)(k0 + 2 * hi) * Mn + (m0 + lo);

#pragma unroll 8
    for (int kk = 0; kk < KC; kk += 4) {
      v2f a;
      a.x = __builtin_nontemporal_load(&Ua[(size_t)kk * Mn]);       // K=kk+2hi
      a.y = __builtin_nontemporal_load(&Ua[(size_t)kk * Mn + Mn]);  // K=kk+2hi+1

      const int rp = (kk >> 1) + hi;  // row-pair index of (n, n+1), n = kk+2hi
      const v2f b0 = *(const v2f*)&xs[rp * RP + 2 * lo];        // f = lo
      const v2f b1 = *(const v2f*)&xs[rp * RP + 32 + 2 * lo];   // f = lo+16

      acc0 = __builtin_amdgcn_wmma_f32_16x16x4_f32(
          false, a, false, b0, (short)0, acc0, false, false);
      acc1 = __builtin_amdgcn_wmma_f32_16x16x4_f32(
          false, a, false, b1, (short)0, acc1, false, false);
    }
  }

  // ---- epilogue: VGPR r holds M = m0 + r + 8*hi, lane lo holds F column ----
  float* __restrict__ ob =
      out + (size_t)b * Mn * Fn + (size_t)(m0 + 8 * hi) * Fn + lo;
#pragma unroll
  for (int r = 0; r < 8; ++r) {
    __builtin_nontemporal_store(acc0[r], &ob[(size_t)r * Fn]);
    __builtin_nontemporal_store(acc1[r], &ob[(size_t)r * Fn + 16]);
  }
}

extern "C" void kernel_launch(void* const* d_in, const int* in_sizes, int n_in,
                              void* d_out, int out_size, void* d_ws, size_t ws_size,
                              hipStream_t stream) {
  const float* x = (const float*)d_in[0];  // [B*N, F] fp32
  const float* U = (const float*)d_in[1];  // [B, N, M] fp32
  float* out     = (float*)d_out;          // [B*M, F] fp32

  dim3 grid(Bn * (Mn / 128));  // 512 workgroups
  dim3 block(256);             // 8 waves (wave32)
  haarpool_wmma_f32<<<grid, block, 0, stream>>>(x, U, out);
}